// FeaturelessRaggedConvolution_45612552683659
// MI455X (gfx1250) — compile-verified
//
#include <hip/hip_runtime.h>
#include <hip/hip_bf16.h>

typedef float v2f __attribute__((ext_vector_type(2)));
typedef float v8f __attribute__((ext_vector_type(8)));

#ifndef FRC_NODES_UDIM
#define FRC_NODES_UDIM 64
#endif

// ---------------------------------------------------------------------------
// Kernel 0: zero the accumulators (num lives in d_out, den lives in d_ws).
// ---------------------------------------------------------------------------
__global__ void frc_zero_kernel(float* __restrict__ num, int n_num,
                                float* __restrict__ den, int n_den) {
    int i = blockIdx.x * blockDim.x + threadIdx.x;
    if (i < n_num) num[i] = 0.0f;
    if (i < n_den) den[i] = 0.0f;
}

// ---------------------------------------------------------------------------
// Kernel 1: per-wave WMMA GEMM (16 edges x 64 u, K=16) + relu + weighted
// atomic scatter into num/den.
//
// A tile (16x4 f32): lane L -> M = L&15, VGPR j + half(L>>4) -> K = j + 2*half
// B tile (4x16 f32): lane L -> N = L&15, VGPR j + half      -> K = j + 2*half
// C tile (16x16 f32): VGPR r, lane L -> M = r + 8*half, N = L&15
// ---------------------------------------------------------------------------
__global__ __launch_bounds__(256) void frc_edge_gemm_scatter(
    const float* __restrict__ emb,   // (64, 16) row-major
    const float* __restrict__ cf,    // (E, 16) row-major
    const float* __restrict__ w,     // (E)
    const int*   __restrict__ idx,   // (E)
    float* __restrict__ num,         // (NODES, 64) accumulator == d_out
    float* __restrict__ den,         // (NODES)
    int E) {
    const int lane  = threadIdx.x & 31;
    const int half  = lane >> 4;     // 0: lanes 0-15, 1: lanes 16-31
    const int l16   = lane & 15;

    const int wave   = (blockIdx.x * blockDim.x + threadIdx.x) >> 5;
    const int nwaves = (gridDim.x * blockDim.x) >> 5;

    // Load B = embedding^T once: b[ut][kc] covers u-tile ut (16 cols of U),
    // k-chunk kc (4 of the 16 F values). Element (K = kc*4 + j + 2*half,
    // N = l16) = emb[(ut*16 + l16)*16 + kc*4 + 2*half + j], j = 0,1.
    v2f b[4][4];
#pragma unroll
    for (int ut = 0; ut < 4; ++ut) {
#pragma unroll
        for (int kc = 0; kc < 4; ++kc) {
            const float* p = emb + (ut * 16 + l16) * 16 + kc * 4 + 2 * half;
            b[ut][kc] = *(const v2f*)p;
        }
    }

    const int ntiles = E >> 4;  // 16 edges per tile
    for (int t = wave; t < ntiles; t += nwaves) {
        const int ebase = t << 4;

        // A tiles: lane row M = l16 -> edge ebase + l16.
        v2f a[4];
        const float* arow = cf + (size_t)(ebase + l16) * 16;
#pragma unroll
        for (int kc = 0; kc < 4; ++kc) {
            a[kc] = *(const v2f*)(arow + kc * 4 + 2 * half);
        }

        // Weights / element offsets for the 8 edge-rows this lane owns in C.
        // Indices are in [0, NODES): 32-bit unsigned offsets are exact and let
        // the backend use SGPR-base + u32-VGPR-offset global addressing.
        float    wr[8];
        unsigned ur[8];
#pragma unroll
        for (int r = 0; r < 8; ++r) {
            const int e = ebase + r + 8 * half;
            wr[r] = w[e];
            ur[r] = (unsigned)idx[e] * (unsigned)FRC_NODES_UDIM + (unsigned)l16;
        }

#pragma unroll
        for (int ut = 0; ut < 4; ++ut) {
            v8f c = {};
#pragma unroll
            for (int kc = 0; kc < 4; ++kc) {
                // D = A(16x4) x B(4x16) + C, f32 accumulate
                c = __builtin_amdgcn_wmma_f32_16x16x4_f32(
                        /*neg_a=*/false, a[kc],
                        /*neg_b=*/false, b[ut][kc],
                        /*c_mod=*/(short)0, c,
                        /*reuse_a=*/false, /*reuse_b=*/false);
            }
            // Scale-then-relu (weights > 0, so relu(x)*w == relu(x*w)); the
            // mul canonicalizes, so the max lowers to a single v_max_num_f32.
            // Scatter: element (M = r + 8*half, N = l16).
#pragma unroll
            for (int r = 0; r < 8; ++r) {
                const float v = fmaxf(c[r] * wr[r], 0.0f);
                unsafeAtomicAdd(num + (ur[r] + (unsigned)(ut * 16)), v);
            }
        }

        // Denominator: one atomic per edge, lanes 0-15 (after all WMMAs).
        if (half == 0) {
            const int e = ebase + l16;
            unsafeAtomicAdd(den + (unsigned)idx[e], w[e]);
        }
    }
}

// ---------------------------------------------------------------------------
// Kernel 2: out[n][u] = num[n][u] / den[n]  (in place on d_out)
// ---------------------------------------------------------------------------
__global__ void frc_divide_kernel(float* __restrict__ out,
                                  const float* __restrict__ den, int n) {
    int i = blockIdx.x * blockDim.x + threadIdx.x;
    if (i < n) {
        out[i] = out[i] / den[i >> 6];  // 64 u-features per node
    }
}

extern "C" void kernel_launch(void* const* d_in, const int* in_sizes, int n_in,
                              void* d_out, int out_size, void* d_ws, size_t ws_size,
                              hipStream_t stream) {
    const float* emb = (const float*)d_in[0];  // (64, 16)
    const float* cf  = (const float*)d_in[1];  // (E, 16)
    const float* w   = (const float*)d_in[2];  // (E)
    const int*   idx = (const int*)d_in[3];    // (E)

    float* out = (float*)d_out;                // (NODES, 64) == num accumulator
    float* den = (float*)d_ws;                 // (NODES)

    const int E     = in_sizes[2];
    const int nodes = out_size / FRC_NODES_UDIM;

    // Zero accumulators (runs every call; harness poisons buffers).
    frc_zero_kernel<<<(out_size + 255) / 256, 256, 0, stream>>>(out, out_size, den, nodes);

    // Main fused GEMM + scatter. 4096 blocks x 8 waves = 32768 waves over
    // E/16 = 200000 edge-tiles (~6 tiles/wave, grid-stride).
    frc_edge_gemm_scatter<<<4096, 256, 0, stream>>>(emb, cf, w, idx, out, den, E);

    // Final normalize.
    frc_divide_kernel<<<(out_size + 255) / 256, 256, 0, stream>>>(out, den, out_size);
}